// LPSolverADMM_56908316672691
// MI455X (gfx1250) — compile-verified
//
#include <hip/hip_runtime.h>
#include <hip/hip_bf16.h>
#include <math.h>

// ---------------------------------------------------------------------------
// ADMM LP solver (fixed 8 outer x 10 PCG iterations), f32.
// A is 8192x4096 f32 (134MB) -> resident in MI455X 192MB L2 after first pass.
// GEMV A*v    : wave-per-row float4 streaming + wave32 shuffle reduce (VALU).
// GEMV A^T*u  : V_WMMA_F32_16X16X4_F32, u broadcast in A-operand row 0,
//               4x16 tile of A as B-operand (contiguous 64B row segments),
//               split-K=16 partials + deterministic combine.
//               Templated on compile-time column stride -> immediate-offset
//               loads + single pointer increment in the hot loop.
// ---------------------------------------------------------------------------

typedef __attribute__((ext_vector_type(2))) float v2f;
typedef __attribute__((ext_vector_type(8))) float v8f;

#define TPB 256
#define SPLITK 16

// ------------------------- setup kernels -----------------------------------

__global__ void colnorm_partial(const float* __restrict__ A, float* __restrict__ part,
                                int m, int n) {
  int j = blockIdx.x * blockDim.x + threadIdx.x;
  int s = blockIdx.y;
  if (j >= n) return;
  int rows = m / SPLITK;
  const float* Ap = A + (size_t)s * rows * n + j;
  float acc = 0.f;
  for (int i = 0; i < rows; ++i) {
    float a = Ap[(size_t)i * n];
    acc += a * a;
  }
  part[(size_t)s * n + j] = acc;
}

__global__ void mdiag_combine(const float* __restrict__ part,
                              const float* __restrict__ sigma_log,
                              const float* __restrict__ rho,
                              float* __restrict__ Mdiag, int n) {
  int j = blockIdx.x * blockDim.x + threadIdx.x;
  if (j >= n) return;
  float s = 0.f;
  for (int k = 0; k < SPLITK; ++k) s += part[(size_t)k * n + j];
  Mdiag[j] = expf(sigma_log[0]) + rho[0] * s;
}

__global__ void setup_vectors(const float* __restrict__ c, const float* __restrict__ d,
                              const float* __restrict__ e, const float* __restrict__ lb,
                              const float* __restrict__ ub,
                              const float* __restrict__ gamma_c, const float* __restrict__ gamma_b,
                              const float* __restrict__ gamma_c_mul, const float* __restrict__ gamma_b_mul,
                              const int* __restrict__ m_ub_p,
                              float* __restrict__ c_s, float* __restrict__ lb_s, float* __restrict__ ub_s,
                              float* __restrict__ x, float* __restrict__ xt,
                              float* __restrict__ z, float* __restrict__ y,
                              int m, int n) {
  int i = blockIdx.x * blockDim.x + threadIdx.x;
  if (i >= m) return;
  float gb = gamma_b_mul[0] * gamma_b[0];
  int m_ub = m_ub_p[0];
  lb_s[i] = lb[i] * ((i >= m_ub) ? gb * e[i] : 1.0f);
  ub_s[i] = ub[i] * ((i < m - n) ? gb * e[i] : 1.0f);
  z[i] = 0.f;
  y[i] = 0.f;
  if (i < n) {
    float gc = gamma_c_mul[0] * gamma_c[0];
    c_s[i] = gc * d[i] * c[i];
    x[i] = 0.f;
    xt[i] = 0.f;
  }
}

// ------------------------- GEMV: u = A * v (m rows) -------------------------

__global__ void gemv_av(const float* __restrict__ A, const float* __restrict__ v,
                        float* __restrict__ out, int m, int n) {
  int row  = blockIdx.x * (blockDim.x >> 5) + (threadIdx.x >> 5);
  int lane = threadIdx.x & 31;
  if (row >= m) return;
  const float4* Arow = (const float4*)(A + (size_t)row * n);
  const float4* v4   = (const float4*)v;
  int n4 = n >> 2;
  float acc = 0.f;
  for (int k = lane; k < n4; k += 32) {
    float4 a = Arow[k];
    float4 b = v4[k];
    acc += a.x * b.x + a.y * b.y + a.z * b.z + a.w * b.w;
  }
  for (int off = 16; off > 0; off >>= 1) acc += __shfl_down(acc, off, 32);
  if (lane == 0) out[row] = acc;
}

// ---------------- GEMV^T via WMMA: part[s][j] = sum_{i in split s} A[i][j]*u[i]
// D(16x16) = Am(16x4) x Bm(4x16); Am row0 = u[k..k+3] (rest 0);
// Bm = A[k..k+3][jt*16 .. jt*16+15]  (4 contiguous 64B row segments).
// A/B-operand layout: lane group g = lane/16 selects K = vgpr + 2*g.
// NC > 0: compile-time column stride (immediate-offset loads); NC==0: runtime.

template <int NC>
__global__ void gemvT_wmma(const float* __restrict__ A, const float* __restrict__ u,
                           float* __restrict__ part, int m, int n_rt) {
  const int n = (NC > 0) ? NC : n_rt;
  int wave   = blockIdx.x * (blockDim.x >> 5) + (threadIdx.x >> 5);
  int ntiles = n >> 4;
  int jt   = wave % ntiles;
  int s    = wave / ntiles;
  int lane = threadIdx.x & 31;
  int g    = lane >> 4;    // K-group select
  int nn   = lane & 15;    // column within tile
  int col  = (jt << 4) + nn;
  int rows = m / SPLITK;
  int i0   = s * rows;

  const size_t sn = (size_t)n;
  // row (K = 2g) base for this lane; rows K=2g+1, 2g+4, 2g+5 via offsets n,4n,5n
  const float* Arow = A + (size_t)(i0 + 2 * g) * sn + col;
  const float* up   = u + i0 + 2 * g;   // 8-byte aligned (i0, 2g even)
  const bool first  = (nn == 0);

  v8f c0 = {};
  v8f c1 = {};
#pragma unroll 2
  for (int i = 0; i < rows; i += 8) {
    // unconditional aligned vector loads of u, masked into A-operand row 0
    v2f ua0 = *(const v2f*)(up);        // {u[i+2g],   u[i+2g+1]}
    v2f ua1 = *(const v2f*)(up + 4);    // {u[i+2g+4], u[i+2g+5]}
    v2f a0, a1, b0, b1;
    a0.x = first ? ua0.x : 0.f;
    a0.y = first ? ua0.y : 0.f;
    a1.x = first ? ua1.x : 0.f;
    a1.y = first ? ua1.y : 0.f;
    b0.x = Arow[0];
    b0.y = Arow[sn];
    b1.x = Arow[4 * sn];
    b1.y = Arow[5 * sn];
    c0 = __builtin_amdgcn_wmma_f32_16x16x4_f32(false, a0, false, b0, (short)0, c0, false, false);
    c1 = __builtin_amdgcn_wmma_f32_16x16x4_f32(false, a1, false, b1, (short)0, c1, false, false);
    Arow += 8 * sn;
    up   += 8;
  }
  v8f c = c0 + c1;
  if (g == 0) part[(size_t)s * n + col] = c[0];  // D row 0 = column dot results
}

// ------------------------- fused vector kernels -----------------------------

__global__ void compute_t(const float* __restrict__ z, const float* __restrict__ y,
                          const float* __restrict__ rho, float* __restrict__ t, int m) {
  int i = blockIdx.x * blockDim.x + threadIdx.x;
  if (i < m) t[i] = rho[0] * z[i] - y[i];
}

__global__ void rhs_kernel(const float* __restrict__ x, const float* __restrict__ c_s,
                           const float* __restrict__ part, const float* __restrict__ sigma_log,
                           float* __restrict__ rhs, int n) {
  int j = blockIdx.x * blockDim.x + threadIdx.x;
  if (j >= n) return;
  float s = 0.f;
  for (int k = 0; k < SPLITK; ++k) s += part[(size_t)k * n + j];
  rhs[j] = expf(sigma_log[0]) * x[j] - c_s[j] + s;
}

__global__ void pcg_init(const float* __restrict__ xt, const float* __restrict__ rhs,
                         const float* __restrict__ part, const float* __restrict__ Mdiag,
                         const float* __restrict__ sigma_log, const float* __restrict__ rho,
                         float* __restrict__ r, float* __restrict__ zp, float* __restrict__ p,
                         int n) {
  int j = blockIdx.x * blockDim.x + threadIdx.x;
  if (j >= n) return;
  float s = 0.f;
  for (int k = 0; k < SPLITK; ++k) s += part[(size_t)k * n + j];
  float w  = expf(sigma_log[0]) * xt[j] + rho[0] * s;  // matvec(xt)
  float rj = rhs[j] - w;
  r[j]  = rj;
  float zj = rj / Mdiag[j];
  zp[j] = zj;
  p[j]  = zj;
}

__global__ void combine_Ap(const float* __restrict__ p, const float* __restrict__ part,
                           const float* __restrict__ sigma_log, const float* __restrict__ rho,
                           float* __restrict__ Ap, int n) {
  int j = blockIdx.x * blockDim.x + threadIdx.x;
  if (j >= n) return;
  float s = 0.f;
  for (int k = 0; k < SPLITK; ++k) s += part[(size_t)k * n + j];
  Ap[j] = expf(sigma_log[0]) * p[j] + rho[0] * s;
}

__global__ void dot_kernel(const float* __restrict__ a, const float* __restrict__ b,
                           float* __restrict__ out, int n) {
  __shared__ float sm[1024];
  float acc = 0.f;
  for (int i = threadIdx.x; i < n; i += blockDim.x) acc += a[i] * b[i];
  sm[threadIdx.x] = acc;
  __syncthreads();
  for (int s = blockDim.x >> 1; s > 0; s >>= 1) {
    if ((int)threadIdx.x < s) sm[threadIdx.x] += sm[threadIdx.x + s];
    __syncthreads();
  }
  if (threadIdx.x == 0) out[0] = sm[0];
}

__global__ void update_x_r(const float* __restrict__ rz, const float* __restrict__ pap,
                           const float* __restrict__ p, const float* __restrict__ Ap,
                           const float* __restrict__ Mdiag,
                           float* __restrict__ xt, float* __restrict__ r, float* __restrict__ zp,
                           int n) {
  int j = blockIdx.x * blockDim.x + threadIdx.x;
  if (j >= n) return;
  float a = rz[0] / pap[0];
  xt[j] += a * p[j];
  float rj = r[j] - a * Ap[j];
  r[j]  = rj;
  zp[j] = rj / Mdiag[j];
}

__global__ void update_p(const float* __restrict__ rznew, const float* __restrict__ rzold,
                         const float* __restrict__ zp, float* __restrict__ p, int n) {
  int j = blockIdx.x * blockDim.x + threadIdx.x;
  if (j >= n) return;
  float beta = rznew[0] / rzold[0];
  p[j] = zp[j] + beta * p[j];
}

__global__ void outer_x(const float* __restrict__ alpha, const float* __restrict__ xt,
                        float* __restrict__ x, int n) {
  int j = blockIdx.x * blockDim.x + threadIdx.x;
  if (j >= n) return;
  float al = alpha[0];
  x[j] = al * xt[j] + (1.f - al) * x[j];
}

__global__ void outer_zy(const float* __restrict__ alpha, const float* __restrict__ rho,
                         const float* __restrict__ zt, const float* __restrict__ lb_s,
                         const float* __restrict__ ub_s,
                         float* __restrict__ z, float* __restrict__ y, int m) {
  int i = blockIdx.x * blockDim.x + threadIdx.x;
  if (i >= m) return;
  float al = alpha[0], rh = rho[0];
  float zh = al * zt[i] + (1.f - al) * z[i];
  float zn = fminf(fmaxf(zh + y[i] / rh, lb_s[i]), ub_s[i]);
  y[i] += rh * (zh - zn);
  z[i] = zn;
}

__global__ void finalize_kernel(const float* __restrict__ x, const float* __restrict__ d,
                                const float* __restrict__ gamma_b, const float* __restrict__ gamma_b_mul,
                                float* __restrict__ out, int n) {
  int j = blockIdx.x * blockDim.x + threadIdx.x;
  if (j >= n) return;
  float gb = gamma_b_mul[0] * gamma_b[0];
  out[j] = x[j] * d[j] / gb;
}

// ------------------------------ host side -----------------------------------

extern "C" void kernel_launch(void* const* d_in, const int* in_sizes, int n_in,
                              void* d_out, int out_size, void* d_ws, size_t ws_size,
                              hipStream_t stream) {
  const float* A         = (const float*)d_in[0];
  const float* c         = (const float*)d_in[1];
  const float* d         = (const float*)d_in[2];
  const float* e         = (const float*)d_in[3];
  const float* lb        = (const float*)d_in[4];
  const float* ub        = (const float*)d_in[5];
  const float* gamma_c   = (const float*)d_in[6];
  const float* gamma_b   = (const float*)d_in[7];
  const float* rho       = (const float*)d_in[8];
  const float* sigma_log = (const float*)d_in[9];
  const float* alpha     = (const float*)d_in[10];
  const float* gcm       = (const float*)d_in[11];
  const float* gbm       = (const float*)d_in[12];
  const int*   m_ub      = (const int*)d_in[13];

  const int n = in_sizes[1];          // 4096
  const int m = in_sizes[3];          // 8192
  float* out = (float*)d_out;

  // workspace carve-up
  float* w = (float*)d_ws;
  size_t off = 0;
  auto alloc = [&](size_t cnt) { float* p = w + off; off += cnt; return p; };
  float* x     = alloc(n);
  float* xt    = alloc(n);
  float* r     = alloc(n);
  float* p_    = alloc(n);
  float* zp    = alloc(n);
  float* Ap    = alloc(n);
  float* rhs   = alloc(n);
  float* c_s   = alloc(n);
  float* Mdiag = alloc(n);
  float* z     = alloc(m);
  float* y     = alloc(m);
  float* u     = alloc(m);
  float* t     = alloc(m);
  float* lb_s  = alloc(m);
  float* ub_s  = alloc(m);
  float* part  = alloc((size_t)SPLITK * n);
  float* scal  = alloc(8);            // [0],[2]: rz ping-pong; [4]: pAp

  const int gn = (n + TPB - 1) / TPB;
  const int gm = (m + TPB - 1) / TPB;
  const int g_av = (m + 7) / 8;                       // wave-per-row, 8 waves/block
  const int g_tw = ((n >> 4) * SPLITK + 7) / 8;       // WMMA waves, 8 waves/block
  const bool fixed = (n == 4096);

  auto launch_gemvT = [&](const float* vec) {
    if (fixed) gemvT_wmma<4096><<<g_tw, TPB, 0, stream>>>(A, vec, part, m, n);
    else       gemvT_wmma<0>   <<<g_tw, TPB, 0, stream>>>(A, vec, part, m, n);
  };

  // ---- setup (recomputed every call; inputs are never mutated) ----
  colnorm_partial<<<dim3(gn, SPLITK), TPB, 0, stream>>>(A, part, m, n);
  mdiag_combine<<<gn, TPB, 0, stream>>>(part, sigma_log, rho, Mdiag, n);
  setup_vectors<<<gm, TPB, 0, stream>>>(c, d, e, lb, ub, gamma_c, gamma_b, gcm, gbm,
                                        m_ub, c_s, lb_s, ub_s, x, xt, z, y, m, n);

  const int OUTER = 8, PCG = 10;
  for (int it = 0; it < OUTER; ++it) {
    // right = sigma*x - c_s + A^T (rho*z - y)
    compute_t<<<gm, TPB, 0, stream>>>(z, y, rho, t, m);
    launch_gemvT(t);
    rhs_kernel<<<gn, TPB, 0, stream>>>(x, c_s, part, sigma_log, rhs, n);

    // r = rhs - matvec(xt);  zp = r/Mdiag;  p = zp
    gemv_av<<<g_av, TPB, 0, stream>>>(A, xt, u, m, n);
    launch_gemvT(u);
    pcg_init<<<gn, TPB, 0, stream>>>(xt, rhs, part, Mdiag, sigma_log, rho, r, zp, p_, n);

    float* rzo = scal + 0;
    float* rzn = scal + 2;
    dot_kernel<<<1, 1024, 0, stream>>>(r, zp, rzo, n);

    for (int k = 0; k < PCG; ++k) {
      gemv_av<<<g_av, TPB, 0, stream>>>(A, p_, u, m, n);
      launch_gemvT(u);
      combine_Ap<<<gn, TPB, 0, stream>>>(p_, part, sigma_log, rho, Ap, n);
      dot_kernel<<<1, 1024, 0, stream>>>(p_, Ap, scal + 4, n);
      update_x_r<<<gn, TPB, 0, stream>>>(rzo, scal + 4, p_, Ap, Mdiag, xt, r, zp, n);
      dot_kernel<<<1, 1024, 0, stream>>>(r, zp, rzn, n);
      update_p<<<gn, TPB, 0, stream>>>(rzn, rzo, zp, p_, n);
      float* tmp = rzo; rzo = rzn; rzn = tmp;
    }

    // zt = A xt; x, z, y updates
    gemv_av<<<g_av, TPB, 0, stream>>>(A, xt, u, m, n);
    outer_x<<<gn, TPB, 0, stream>>>(alpha, xt, x, n);
    outer_zy<<<gm, TPB, 0, stream>>>(alpha, rho, u, lb_s, ub_s, z, y, m);
  }

  finalize_kernel<<<gn, TPB, 0, stream>>>(x, d, gamma_b, gbm, out, n);
  (void)n_in; (void)out_size; (void)ws_size;
}